// ElementalEmbeddedAtomPotential_53772990546319
// MI455X (gfx1250) — compile-verified
//
#include <hip/hip_runtime.h>
#include <math.h>

typedef __attribute__((ext_vector_type(2))) float v2f;
typedef __attribute__((ext_vector_type(8))) float v8f;

#define NBASIS 128
#define NCHUNK 32                       // 128 K / 4-per-WMMA
#define CUTOFF 6.0f
#define DELTA  (CUTOFF / (NBASIS - 1))  // center spacing
#define GAMMA_ ((NBASIS - 1) / CUTOFF)  // 1 / spacing
#define PI_F   3.14159265358979323846f
#define LOG2E_F 1.44269504088896340736f
// exp(-g*t^2) == exp2(K2 * t^2),  K2 = -g * log2(e)
#define K2_    (-GAMMA_ * LOG2E_F)

#if __has_builtin(__builtin_amdgcn_exp2f)
#define FAST_EXP2(x) __builtin_amdgcn_exp2f(x)
#else
#define FAST_EXP2(x) exp2f(x)
#endif

__device__ __forceinline__ float softplus_(float x) {
  // logaddexp(x, 0) = max(x,0) + log1p(exp(-|x|))
  return fmaxf(x, 0.0f) + log1pf(expf(-fabsf(x)));
}

__device__ __forceinline__ float blockReduceSum256(float v) {
  __shared__ float sred[256];
  sred[threadIdx.x] = v;
  __syncthreads();
  for (int s = blockDim.x >> 1; s > 0; s >>= 1) {
    if (threadIdx.x < (unsigned)s) sred[threadIdx.x] += sred[threadIdx.x + s];
    __syncthreads();
  }
  return sred[0];
}

// ---------------------------------------------------------------------------
// Kernel 0: zero node-density scratch and the forces output region
// ---------------------------------------------------------------------------
__global__ void eam_zero(float* __restrict__ a, int na, float* __restrict__ b, int nb) {
  const int i = blockIdx.x * blockDim.x + threadIdx.x;
  const int stride = gridDim.x * blockDim.x;
  for (int j = i; j < na; j += stride) a[j] = 0.0f;
  for (int j = i; j < nb; j += stride) b[j] = 0.0f;
}

// ---------------------------------------------------------------------------
// Kernel 1: per-edge forward pass.
//   For each edge: d = |r|, then via WMMA f32 16x16x4 accumulate
//     SA = sum_k g_k * sp_k        SB = sum_k g_k * c_k * sp_k
//     SC = sum_k g_k * pp_k        SD = sum_k g_k * c_k * pp_k
//   with g_k = exp(-gamma (d - c_k)^2), sp = softplus(phi_density), pp = phi_pair.
//   A operand = 16(ch) x 4(k) coefficient tile (rows 0..3 nonzero), constant
//   across edge tiles -> hoisted into VGPRs. B operand = g, computed on the
//   fly with packed-f32 argument math + bare v_exp_f32 (exp2 with folded
//   gamma*log2e), co-executing with the matrix pipe.
//   D layout: lane n = edge n, VGPR m = channel m.
// ---------------------------------------------------------------------------
__global__ __launch_bounds__(256) void eam_edge_fwd(
    const float* __restrict__ r, const int* __restrict__ dst,
    const float* __restrict__ phiD, const float* __restrict__ phiP,
    float* __restrict__ local, float* __restrict__ gd_out, float* __restrict__ gp_out,
    float* __restrict__ partial_pair, int E)
{
  __shared__ v2f sW[64 * 16];  // [k-pair][channel] -> {W[ch][2kp], W[ch][2kp+1]}

  for (int idx = threadIdx.x; idx < 64 * 16; idx += blockDim.x) {
    const int kp = idx >> 4, ch = idx & 15;
    const int k0 = 2 * kp, k1 = k0 + 1;
    const float c0 = k0 * DELTA, c1 = k1 * DELTA;
    v2f w = {0.0f, 0.0f};
    if (ch < 4) {
      float a0, a1;
      if (ch < 2) { a0 = softplus_(phiD[k0]); a1 = softplus_(phiD[k1]); }
      else        { a0 = phiP[k0];            a1 = phiP[k1]; }
      if (ch & 1) { a0 *= c0; a1 *= c1; }
      w.x = a0; w.y = a1;
    }
    sW[idx] = w;
  }
  __syncthreads();

  const int lane = threadIdx.x & 31;   // wave32
  const int col  = lane & 15;          // edge slot / coeff channel row
  const int hi   = lane >> 4;          // K half: lanes 0-15 -> k{0,1}, 16-31 -> k{2,3}

  // Hoist the A operand (coefficients) into registers: invariant across tiles.
  v2f aReg[NCHUNK];
#pragma unroll
  for (int t = 0; t < NCHUNK; ++t)
    aReg[t] = sW[(2 * t + hi) * 16 + col];

  const int wavesPerBlk = blockDim.x >> 5;
  const int wave   = blockIdx.x * wavesPerBlk + (threadIdx.x >> 5);
  const int nWaves = gridDim.x * wavesPerBlk;
  const int nTiles = (E + 15) >> 4;

  float pairAcc = 0.0f;

  for (int tile = wave; tile < nTiles; tile += nWaves) {
    const int e = tile * 16 + col;
    const bool valid = (e < E);

    // prefetch next tile's coordinates
    const int eNext = (tile + nWaves) * 16 + col;
    if (eNext < E) __builtin_prefetch(&r[3 * eNext], 0, 0);

    float x = 0.0f, y = 0.0f, z = 0.0f;
    if (valid) { x = r[3 * e]; y = r[3 * e + 1]; z = r[3 * e + 2]; }
    const float dist = sqrtf(x * x + y * y + z * z);

    v8f acc = {0.f, 0.f, 0.f, 0.f, 0.f, 0.f, 0.f, 0.f};
    const float base = dist - (2 * hi) * DELTA;      // dist - c_{2hi}
    const v2f base2 = {base, base - DELTA};          // k and k+1 offsets
#pragma unroll
    for (int t = 0; t < NCHUNK; ++t) {
      // tt = dist - c_{4t+2hi+{0,1}} : one packed add (splat of chunk const)
      const v2f tt  = base2 - (float)(4 * t) * DELTA;
      // arg = K2 * tt^2 : two packed muls
      const v2f arg = (tt * tt) * (float)K2_;
      v2f g;
      g.x = FAST_EXP2(arg.x);
      g.y = FAST_EXP2(arg.y);
      acc = __builtin_amdgcn_wmma_f32_16x16x4_f32(
          false, aReg[t], false, g, (short)0, acc, false, false);
    }

    if (lane < 16 && valid) {
      const float SA = acc[0], SB = acc[1], SC = acc[2], SD = acc[3];
      float sn, cs;
      __sincosf(dist * (PI_F / CUTOFF), &sn, &cs);
      const float phi  = 0.25f * (1.0f + cs);              // 0.5 * fcut
      const float phip = -0.25f * (PI_F / CUTOFF) * sn;    // d(0.5*fcut)/dd
      const float invd = 1.0f / dist;

      const float dens   = phi * SA;
      const float ddens  = phi * (-2.0f * GAMMA_) * (dist * SA - SB) + phip * SA;
      const float pairn  = phi * SC;                       // pair numerator
      const float dpairn = phi * (-2.0f * GAMMA_) * (dist * SC - SD) + phip * SC;
      const float dpair  = (dpairn - pairn * invd) * invd; // d(pairn/d)/dd

      atomicAdd(&local[dst[e]], dens);
      gd_out[e] = ddens * invd;   // * (r/d) later
      gp_out[e] = dpair * invd;
      pairAcc += pairn * invd;    // pair energy contribution
    }
  }

  const float bsum = blockReduceSum256(pairAcc);
  if (threadIdx.x == 0) partial_pair[blockIdx.x] = bsum;
}

// ---------------------------------------------------------------------------
// Kernel 2: per-node embedding energy F(local) and derivative F'(local).
//   POWERS=[0.5,1,2,3,4], SF=[2,1,1/2,1/6,1/24]
// ---------------------------------------------------------------------------
__global__ __launch_bounds__(256) void eam_node(
    const float* __restrict__ local, const float* __restrict__ w,
    float* __restrict__ Fp, float* __restrict__ partial_F, int N)
{
  const float w0 = w[0], w1 = w[1], w2 = w[2], w3 = w[3], w4 = w[4];
  float Facc = 0.0f;
  for (int n = blockIdx.x * blockDim.x + threadIdx.x; n < N;
       n += gridDim.x * blockDim.x) {
    const float x  = local[n];
    const float sq = sqrtf(x);
    const float x2 = x * x;
    const float F  = 2.0f * w0 * sq + w1 * x + 0.5f * w2 * x2 +
                     (1.0f / 6.0f) * w3 * x2 * x + (1.0f / 24.0f) * w4 * x2 * x2;
    const float Fd = w0 / sq + w1 + w2 * x + 0.5f * w3 * x2 +
                     (1.0f / 6.0f) * w4 * x2 * x;
    Facc += F;
    Fp[n] = Fd;
  }
  const float bsum = blockReduceSum256(Facc);
  if (threadIdx.x == 0) partial_F[blockIdx.x] = bsum;
}

// ---------------------------------------------------------------------------
// Kernel 3: forces.  dE/dr_e = (F'(local[dst]) * gd + gp) * r_e ;
//   forces[src] += dE/dr, forces[dst] -= dE/dr.
// ---------------------------------------------------------------------------
__global__ __launch_bounds__(256) void eam_force(
    const float* __restrict__ r, const int* __restrict__ src,
    const int* __restrict__ dst, const float* __restrict__ gd,
    const float* __restrict__ gp, const float* __restrict__ Fp,
    float* __restrict__ forces, int E)
{
  for (int e = blockIdx.x * blockDim.x + threadIdx.x; e < E;
       e += gridDim.x * blockDim.x) {
    const int s = src[e], d = dst[e];
    const float g = Fp[d] * gd[e] + gp[e];
    const float fx = g * r[3 * e], fy = g * r[3 * e + 1], fz = g * r[3 * e + 2];
    atomicAdd(&forces[3 * s + 0],  fx);
    atomicAdd(&forces[3 * s + 1],  fy);
    atomicAdd(&forces[3 * s + 2],  fz);
    atomicAdd(&forces[3 * d + 0], -fx);
    atomicAdd(&forces[3 * d + 1], -fy);
    atomicAdd(&forces[3 * d + 2], -fz);
  }
}

// ---------------------------------------------------------------------------
// Kernel 4: deterministic final energy reduction over block partials.
// ---------------------------------------------------------------------------
__global__ __launch_bounds__(256) void eam_energy(
    const float* __restrict__ partial_pair, int n1,
    const float* __restrict__ partial_F, int n2,
    float* __restrict__ energy)
{
  float v = 0.0f;
  for (int i = threadIdx.x; i < n1; i += blockDim.x) v += partial_pair[i];
  for (int i = threadIdx.x; i < n2; i += blockDim.x) v += partial_F[i];
  const float s = blockReduceSum256(v);
  if (threadIdx.x == 0) energy[0] = s;
}

// ---------------------------------------------------------------------------
extern "C" void kernel_launch(void* const* d_in, const int* in_sizes, int n_in,
                              void* d_out, int out_size, void* d_ws, size_t ws_size,
                              hipStream_t stream) {
  const float* r    = (const float*)d_in[0];
  const int*   src  = (const int*)  d_in[1];
  const int*   dst  = (const int*)  d_in[2];
  const float* phiD = (const float*)d_in[3];
  const float* phiP = (const float*)d_in[4];
  const float* emb  = (const float*)d_in[5];

  const int E = in_sizes[1];            // 800000
  const int N = (out_size - 1) / 3;     // 50000

  float* out    = (float*)d_out;
  float* energy = out;                  // d_out[0]
  float* forces = out + 1;              // d_out[1 .. 3N]

  // workspace layout (floats): gd[E], gp[E], local[N], Fp[N], partials
  float* ws    = (float*)d_ws;
  float* gd    = ws;
  float* gp    = gd + E;
  float* local = gp + E;
  float* Fp    = local + N;
  const int NB1 = 512;   // blocks for edge-forward kernel
  const int NB2 = 128;   // blocks for node kernel
  float* ppair = Fp + N;
  float* pF    = ppair + NB1;

  eam_zero    <<<256,  256, 0, stream>>>(local, N, forces, 3 * N);
  eam_edge_fwd<<<NB1,  256, 0, stream>>>(r, dst, phiD, phiP, local, gd, gp, ppair, E);
  eam_node    <<<NB2,  256, 0, stream>>>(local, emb, Fp, pF, N);
  eam_force   <<<1024, 256, 0, stream>>>(r, src, dst, gd, gp, Fp, forces, E);
  eam_energy  <<<1,    256, 0, stream>>>(ppair, NB1, pF, NB2, energy);
}